// EMAVQ_1503238553778
// MI455X (gfx1250) — compile-verified
//
#include <hip/hip_runtime.h>
#include <hip/hip_bf16.h>

typedef __attribute__((ext_vector_type(2))) float v2f;
typedef __attribute__((ext_vector_type(8))) float v8f;

#define VQ_VOCAB 8192
#define VQ_D     3
#define VQ_N     16384            // B*L = 4*4096
#define VQ_BETA  0.25f
#define VQ_SEGS  4                // codebook split: waves per point-tile
#define VQ_TILES (VQ_VOCAB / 16)  // 512 code tiles
#define VQ_TSEG  (VQ_TILES / VQ_SEGS)

// ---------------------------------------------------------------------------
// Kernel 1: pack codebook rows as (e0, e1, e2, ||e||^2) and zero the loss slot.
// A-fragment becomes a single b64 load per lane:
//   lanes 0-15  load (e0,e1)      = K0,K1
//   lanes 16-31 load (e2,||e||^2) = K2,K3   (ISA 32-bit A 16x4 layout)
// ---------------------------------------------------------------------------
__global__ void vq_pack_codebook(const float* __restrict__ emb,
                                 float4* __restrict__ packed,
                                 float* __restrict__ loss_slot) {
  int c = blockIdx.x * blockDim.x + threadIdx.x;
  if (c == 0) *loss_slot = 0.0f;
  if (c < VQ_VOCAB) {
    float x = emb[c * 3 + 0];
    float y = emb[c * 3 + 1];
    float z = emb[c * 3 + 2];
    packed[c] = make_float4(x, y, z, x * x + y * y + z * z);
  }
}

// ---------------------------------------------------------------------------
// Kernel 2: 256-thread blocks = 8 waves = 2 point-tiles x 4 codebook segments.
// Hot loop per code tile (branch-free, EXEC-uniform):
//   1 b64 load (4-deep pipelined) + 1 v_wmma_f32_16x16x4_f32
//   + 7-op min tree + cmp + 2 cndmask.
// Index recovery + cross-segment merge happen once, in the epilogue.
// ---------------------------------------------------------------------------
__launch_bounds__(256)
__global__ void vq_argmin_wmma(const float* __restrict__ feats,
                               const float* __restrict__ emb,
                               const float* __restrict__ packed,   // [VOCAB*4]
                               float* __restrict__ out_quant,      // [N*3]
                               float* __restrict__ out_idx,        // [N]
                               float* __restrict__ out_loss) {     // [1]
  __shared__ float s_val[2][16][VQ_SEGS];
  __shared__ int   s_idx[2][16][VQ_SEGS];

  const int lane = threadIdx.x & 31;
  const int wave = threadIdx.x >> 5;     // 0..7
  const int ptl  = wave >> 2;            // local point-tile 0..1
  const int seg  = wave & (VQ_SEGS - 1); // codebook segment 0..3
  const int grp  = lane >> 4;            // 0: K0/K1 half, 1: K2/K3 half
  const int ln   = lane & 15;
  const int tile = blockIdx.x * 2 + ptl; // global 16-point tile
  const int p    = tile * 16 + ln;       // this lane's point

  const float x0 = feats[p * 3 + 0];
  const float x1 = feats[p * 3 + 1];
  const float x2 = feats[p * 3 + 2];

  // B-fragment (4x16): column n = (-2x0, -2x1, -2x2, 1). Branchless per half.
  v2f bfrag;
  bfrag[0] = grp ? (-2.0f * x2) : (-2.0f * x0);
  bfrag[1] = grp ? 1.0f         : (-2.0f * x1);

  float best  = __builtin_inff();
  int   bestT = seg * VQ_TSEG;           // code tile holding the min

  const int tBeg = seg * VQ_TSEG, tEnd = tBeg + VQ_TSEG;

  // A-fragment stream: tile d (relative) lives at ap + d*64 floats (256 B).
  const float* ap = packed + (size_t)tBeg * 64 + ln * 4 + grp * 2;

  // 4-deep software pipeline: 4 loads always in flight.
  v2f a[4];
  #pragma unroll
  for (int k = 0; k < 4; ++k) a[k] = *(const v2f*)(ap + k * 64);

  int t = tBeg;
  for (; t < tEnd - 4; t += 4, ap += 4 * 64) {
    #pragma unroll
    for (int k = 0; k < 4; ++k) {
      v8f acc = {};
      acc = __builtin_amdgcn_wmma_f32_16x16x4_f32(
          false, a[k], false, bfrag, (short)0, acc, false, false);
      a[k] = *(const v2f*)(ap + (k + 4) * 64);    // refill slot k (tile t+k+4)
      float m0 = fminf(fminf(acc[0], acc[1]), fminf(acc[2], acc[3]));
      float m1 = fminf(fminf(acc[4], acc[5]), fminf(acc[6], acc[7]));
      float tm = fminf(m0, m1);
      if (tm < best) { best = tm; bestT = t + k; }
    }
  }
  // Drain the pipeline (last 4 tiles).
  #pragma unroll
  for (int k = 0; k < 4; ++k) {
    v8f acc = {};
    acc = __builtin_amdgcn_wmma_f32_16x16x4_f32(
        false, a[k], false, bfrag, (short)0, acc, false, false);
    float m0 = fminf(fminf(acc[0], acc[1]), fminf(acc[2], acc[3]));
    float m1 = fminf(fminf(acc[4], acc[5]), fminf(acc[6], acc[7]));
    float tm = fminf(m0, m1);
    if (tm < best) { best = tm; bestT = t + k; }
  }

  // ---- Epilogue: recover the index within the winning tile (scalar, once).
  const float m2x0 = -2.0f * x0, m2x1 = -2.0f * x1, m2x2 = -2.0f * x2;
  float bv = __builtin_inff();
  int   bi = 0;
  {
    const int base = bestT * 16 + grp * 8;
    #pragma unroll
    for (int j = 0; j < 8; ++j) {
      const float4 e = ((const float4*)packed)[base + j];
      float v = fmaf(m2x0, e.x, fmaf(m2x1, e.y, fmaf(m2x2, e.z, e.w)));
      if (v < bv) { bv = v; bi = base + j; }
    }
  }
  // Merge the two lane-halves (codes m%16<8 vs >=8).
  {
    float ob = __shfl_xor(bv, 16, 32);
    int   oi = __shfl_xor(bi, 16, 32);
    if (ob < bv || (ob == bv && oi < bi)) { bv = ob; bi = oi; }
  }
  // Publish per-segment winner.
  if (grp == 0) { s_val[ptl][ln][seg] = bv; s_idx[ptl][ln][seg] = bi; }
  __syncthreads();

  // ---- Cross-segment reduce + outputs: segment-0 waves only.
  float part = 0.0f;
  if (seg == 0 && grp == 0) {
    float fb = s_val[ptl][ln][0];
    int   fi = s_idx[ptl][ln][0];
    #pragma unroll
    for (int s = 1; s < VQ_SEGS; ++s) {
      float v = s_val[ptl][ln][s];
      int   i = s_idx[ptl][ln][s];
      if (v < fb || (v == fb && i < fi)) { fb = v; fi = i; }
    }
    const float q0 = emb[fi * 3 + 0];
    const float q1 = emb[fi * 3 + 1];
    const float q2 = emb[fi * 3 + 2];
    out_quant[p * 3 + 0] = q0;     // STE output == quant numerically
    out_quant[p * 3 + 1] = q1;
    out_quant[p * 3 + 2] = q2;
    out_idx[p] = (float)fi;
    const float d0 = q0 - x0, d1 = q1 - x1, d2 = q2 - x2;
    part = d0 * d0 + d1 * d1 + d2 * d2;
  }
  // vq_loss = (1+beta) * mean((quant-feats)^2); wave-reduce, 1 atomic per
  // segment-0 wave (2 per block).
  #pragma unroll
  for (int s = 16; s >= 1; s >>= 1) part += __shfl_xor(part, s, 32);
  if (lane == 0 && seg == 0)
    atomicAdd(out_loss, part * (1.0f + VQ_BETA) / (float)(VQ_N * VQ_D));
}

// ---------------------------------------------------------------------------
extern "C" void kernel_launch(void* const* d_in, const int* in_sizes, int n_in,
                              void* d_out, int out_size, void* d_ws, size_t ws_size,
                              hipStream_t stream) {
  const float* feats = (const float*)d_in[0];   // [4,4096,3] f32
  const float* emb   = (const float*)d_in[1];   // [8192,3]   f32

  float* out       = (float*)d_out;
  float* out_quant = out;                       // N*3 = 49152
  float* out_idx   = out + VQ_N * 3;            // N   = 16384
  float* out_loss  = out + out_size - 1;        // scalar loss (last slot)

  float* packed = (float*)d_ws;                 // VOCAB*4 floats = 128 KB

  vq_pack_codebook<<<(VQ_VOCAB + 255) / 256, 256, 0, stream>>>(
      emb, (float4*)packed, out_loss);

  // 2 point-tiles per block, 4 segment-waves each -> 4096 waves.
  const int blocks = (VQ_N / 16) / 2;           // 512
  vq_argmin_wmma<<<blocks, 256, 0, stream>>>(
      feats, emb, packed, out_quant, out_idx, out_loss);
}